// TensorVolVM_44848048504943
// MI455X (gfx1250) — compile-verified
//
#include <hip/hip_runtime.h>

#define N_COMP 48
#define RES    300
#define N_FEATO 27
#define N_PTS  1000000
#define KDIM   (3 * N_COMP)   // 144

typedef __attribute__((ext_vector_type(2))) float v2f;
typedef __attribute__((ext_vector_type(8))) float v8f;

__launch_bounds__(128, 2)
__global__ void tensorvm_kernel(const float* __restrict__ xyz,
                                const float* __restrict__ planes,
                                const float* __restrict__ lines,
                                const float* __restrict__ basisW,
                                float* __restrict__ out)
{
    const int lane = threadIdx.x & 31;
    const int wave = threadIdx.x >> 5;
    const int tile = blockIdx.x * 4 + wave;     // 62500 tiles exactly

    const int mrow = lane & 15;                 // point row within tile
    const int hi   = lane >> 4;                 // K-offset half: {0,1} vs {2,3}

    const int p = tile * 16 + mrow;             // global point id (always in range)

    // --- per-point coordinates ---
    const float px = xyz[p * 3 + 0];
    const float py = xyz[p * 3 + 1];
    const float pz = xyz[p * 3 + 2];
    const float g[3] = {px, py, pz};

    // PLANE_AX = ((2,1),(2,0),(1,0)): gx = xyz[a], gy = xyz[b]; line uses xyz[i]
    const int AX[3] = {2, 2, 1};
    const int BX[3] = {1, 0, 0};

    int   x0i[3], x1i[3], y0i[3], y1i[3], l0i[3], l1i[3];
    float wxv[3], wyv[3], lwv[3];

    #pragma unroll
    for (int i = 0; i < 3; ++i) {
        const float scale = 0.5f * (float)(RES - 1);
        float x = (g[AX[i]] + 1.0f) * scale;
        float y = (g[BX[i]] + 1.0f) * scale;
        float x0f = floorf(x), y0f = floorf(y);
        wxv[i] = x - x0f;
        wyv[i] = y - y0f;
        int x0 = (int)fminf(fmaxf(x0f, 0.0f), (float)(RES - 1));
        int y0 = (int)fminf(fmaxf(y0f, 0.0f), (float)(RES - 1));
        x0i[i] = x0;
        y0i[i] = y0;
        x1i[i] = min(x0 + 1, RES - 1);
        y1i[i] = min(y0 + 1, RES - 1);

        float ly  = (g[i] + 1.0f) * scale;
        float l0f = floorf(ly);
        lwv[i] = ly - l0f;
        int l0 = (int)fminf(fmaxf(l0f, 0.0f), (float)(RES - 1));
        l0i[i] = l0;
        l1i[i] = min(l0 + 1, RES - 1);
    }

    // --- B-column setup (N tiles: cols 0..15 and 16..31, 27..31 masked to 0) ---
    const int   ncol0 = mrow;
    const int   ncol1 = 16 + mrow;
    const float bm1   = (ncol1 < N_FEATO) ? 1.0f : 0.0f;
    const int   brow1 = (ncol1 < N_FEATO) ? ncol1 : (N_FEATO - 1);

    v8f acc0 = {};
    v8f acc1 = {};

    #pragma unroll
    for (int i = 0; i < 3; ++i) {
        const float wx = wxv[i], wy = wyv[i], lw = lwv[i];
        const int pb   = (i * N_COMP) * (RES * RES);     // plane base (fits int32)
        const int row0 = y0i[i] * RES, row1 = y1i[i] * RES;
        const int xa   = x0i[i], xb = x1i[i];
        const int lb   = (i * N_COMP) * RES;             // line base
        const int l0   = l0i[i], l1 = l1i[i];

        for (int j = 0; j < 12; ++j) {
            const int c0 = j * 4 + hi * 2;   // channel pair this lane owns for this K-chunk
            float f[2];
            #pragma unroll
            for (int t = 0; t < 2; ++t) {
                const int c = c0 + t;
                const float* pl = planes + pb + c * (RES * RES);
                float v00 = pl[row0 + xa];
                float v01 = pl[row0 + xb];
                float v10 = pl[row1 + xa];
                float v11 = pl[row1 + xb];
                float bil = (1.0f - wy) * ((1.0f - wx) * v00 + wx * v01)
                          +          wy * ((1.0f - wx) * v10 + wx * v11);
                const float* ln = lines + lb + c * RES;
                float li = (1.0f - lw) * ln[l0] + lw * ln[l1];
                f[t] = bil * li;
            }
            v2f a;
            a.x = f[0];
            a.y = f[1];

            const int k0 = i * N_COMP + c0;  // absolute K index (incl. hi offset)
            v2f b0, b1;
            b0.x = basisW[ncol0 * KDIM + k0];
            b0.y = basisW[ncol0 * KDIM + k0 + 1];
            b1.x = basisW[brow1 * KDIM + k0] * bm1;
            b1.y = basisW[brow1 * KDIM + k0 + 1] * bm1;

            // D = A(16x4 f32) x B(4x16 f32) + C, full fp32 — matches reference precision
            acc0 = __builtin_amdgcn_wmma_f32_16x16x4_f32(
                false, a, false, b0, (short)0, acc0, false, false);
            acc1 = __builtin_amdgcn_wmma_f32_16x16x4_f32(
                false, a, false, b1, (short)0, acc1, false, false);
        }
    }

    // --- store D: VGPR r -> M = r + 8*hi, N = lane&15 (+16 for tile 1) ---
    #pragma unroll
    for (int r = 0; r < 8; ++r) {
        const int M  = r + hi * 8;
        const int pr = tile * 16 + M;
        out[pr * N_FEATO + ncol0] = acc0[r];
        if (ncol1 < N_FEATO)
            out[pr * N_FEATO + ncol1] = acc1[r];
    }
}

extern "C" void kernel_launch(void* const* d_in, const int* in_sizes, int n_in,
                              void* d_out, int out_size, void* d_ws, size_t ws_size,
                              hipStream_t stream) {
    const float* xyz    = (const float*)d_in[0];
    const float* planes = (const float*)d_in[1];
    const float* lines  = (const float*)d_in[2];
    const float* basisW = (const float*)d_in[3];
    float* out = (float*)d_out;

    const int tiles  = N_PTS / 16;   // 62500
    const int blocks = tiles / 4;    // 15625, 4 waves (128 threads) per block
    tensorvm_kernel<<<dim3(blocks), dim3(128), 0, stream>>>(xyz, planes, lines, basisW, out);
}